// EfficientGNN_6004364280337
// MI455X (gfx1250) — compile-verified
//
#include <hip/hip_runtime.h>

typedef __attribute__((ext_vector_type(2))) float v2f;
typedef __attribute__((ext_vector_type(8))) float v8f;

#define NUM_GRAPHS 512

// ---------------------------------------------------------------------------
// Degree: atomic count of in-degree at dst, then dinv = rsqrt(deg+1),
// invdeg = 1/(deg+1)  (self-loop included, matching the reference).
// ---------------------------------------------------------------------------
__global__ void deg_count_kernel(const long long* __restrict__ dst,
                                 float* __restrict__ deg, int E) {
    int e = blockIdx.x * blockDim.x + threadIdx.x;
    if (e < E) atomicAdd(&deg[(int)dst[e]], 1.0f);
}

__global__ void deg_finalize_kernel(float* __restrict__ dinv,
                                    float* __restrict__ invdeg, int N) {
    int n = blockIdx.x * blockDim.x + threadIdx.x;
    if (n < N) {
        float d = dinv[n] + 1.0f;     // dinv slot currently holds raw count
        dinv[n]   = rsqrtf(d);
        invdeg[n] = 1.0f / d;
    }
}

// ---------------------------------------------------------------------------
// Dense GEMM  Out[N x 64] = A[N x 64] @ W[64 x 64]  using f32 WMMA 16x16x4.
// blockDim = 128 (4 wave32s); wave w owns the 16x16 tile at columns [16w,16w+16).
// FUSE_PRE applies relu(a + preBias[k]) to the A operand (layer-1 epilogue
// fused into layer-2's A read).
//
// A-fragment (16x4 f32): lanes 0-15 -> K = {0,1}, lanes 16-31 -> K = {2,3},
// row M = lane % 16. B-fragment (4x16): same K split, column N = lane % 16.
// C/D: VGPR i -> row (i + 8*(lane>=16)), col = lane % 16.
// ---------------------------------------------------------------------------
template <bool FUSE_PRE>
__global__ void gemm64_wmma_kernel(const float* __restrict__ A,
                                   const float* __restrict__ W,
                                   const float* __restrict__ preBias,
                                   float* __restrict__ Out) {
    const int lane    = threadIdx.x & 31;
    const int wave    = threadIdx.x >> 5;
    const int colBase = wave * 16;
    const int m       = lane & 15;
    const int kh      = (lane >> 4) << 1;   // 0 or 2
    const int row0    = blockIdx.x * 16;

    const float* __restrict__ arow = A + (size_t)(row0 + m) * 64;

    v8f acc = {};
#pragma unroll
    for (int k = 0; k < 64; k += 4) {
        float a0 = arow[k + kh];
        float a1 = arow[k + kh + 1];
        if (FUSE_PRE) {
            a0 = fmaxf(a0 + preBias[k + kh], 0.0f);
            a1 = fmaxf(a1 + preBias[k + kh + 1], 0.0f);
        }
        v2f av = {a0, a1};
        v2f bv = {W[(k + kh) * 64 + colBase + m],
                  W[(k + kh + 1) * 64 + colBase + m]};
        acc = __builtin_amdgcn_wmma_f32_16x16x4_f32(
            /*neg_a=*/false, av, /*neg_b=*/false, bv,
            /*c_mod=*/(short)0, acc, /*reuse_a=*/false, /*reuse_b=*/false);
    }

    const int mbase = (lane >> 4) * 8;
#pragma unroll
    for (int i = 0; i < 8; ++i) {
        Out[(size_t)(row0 + mbase + i) * 64 + colBase + m] = acc[i];
    }
}

// ---------------------------------------------------------------------------
// Self-loop contribution, also initializes the scatter accumulator:
//   B[n, :] = A[n, :] * invdeg[n]          (float4 vectorized, N*16 items)
// ---------------------------------------------------------------------------
__global__ void selfloop_init_kernel(const float* __restrict__ A,
                                     const float* __restrict__ invdeg,
                                     float* __restrict__ B, int N16) {
    int idx = blockIdx.x * blockDim.x + threadIdx.x;
    if (idx < N16) {
        float4 a = reinterpret_cast<const float4*>(A)[idx];
        float  s = invdeg[idx >> 4];
        float4 o;
        o.x = a.x * s; o.y = a.y * s; o.z = a.z * s; o.w = a.w * s;
        reinterpret_cast<float4*>(B)[idx] = o;
    }
}

// ---------------------------------------------------------------------------
// Edge scatter (the memory-bound hot loop):
//   for each edge e, feature f:  B[dst,f] += A[src,f] * dinv[src]*dinv[dst]
// Thread mapping: f = idx & 63, e = idx >> 6  -> fully coalesced gathers and
// 64-contiguous atomic targets per edge; dinv/edge-index loads are broadcast.
// ---------------------------------------------------------------------------
__global__ void scatter_kernel(const float* __restrict__ H,
                               const long long* __restrict__ src,
                               const long long* __restrict__ dst,
                               const float* __restrict__ dinv,
                               float* __restrict__ B, long long total) {
    long long idx = (long long)blockIdx.x * blockDim.x + threadIdx.x;
    if (idx < total) {
        long long e = idx >> 6;
        int f = (int)(idx & 63);
        int s = (int)src[e];
        int d = (int)dst[e];
        float w = dinv[s] * dinv[d];
        atomicAdd(&B[(size_t)d * 64 + f], H[(size_t)s * 64 + f] * w);
    }
}

// ---------------------------------------------------------------------------
// Pooling: pooled[batch[n], f] += relu(B2[n,f] + b2[f])
// ---------------------------------------------------------------------------
__global__ void pool_kernel(const float* __restrict__ B2,
                            const float* __restrict__ b2,
                            const long long* __restrict__ batch,
                            float* __restrict__ pooled, int total) {
    int idx = blockIdx.x * blockDim.x + threadIdx.x;
    if (idx < total) {
        int n = idx >> 6, f = idx & 63;
        float v = fmaxf(B2[idx] + b2[f], 0.0f);
        atomicAdd(&pooled[(int)batch[n] * 64 + f], v);
    }
}

// ---------------------------------------------------------------------------
// Head: out[g, c] = pooled[g, :] . Wlin[:, c] + blin[c]   (512 x 2)
// ---------------------------------------------------------------------------
__global__ void final_linear_kernel(const float* __restrict__ pooled,
                                    const float* __restrict__ Wlin,
                                    const float* __restrict__ blin,
                                    float* __restrict__ out, int total) {
    int idx = blockIdx.x * blockDim.x + threadIdx.x;
    if (idx < total) {
        int g = idx >> 1, c = idx & 1;
        float s = blin[c];
#pragma unroll
        for (int k = 0; k < 64; ++k)
            s += pooled[g * 64 + k] * Wlin[k * 2 + c];
        out[idx] = s;
    }
}

// ---------------------------------------------------------------------------
extern "C" void kernel_launch(void* const* d_in, const int* in_sizes, int n_in,
                              void* d_out, int out_size, void* d_ws, size_t ws_size,
                              hipStream_t stream) {
    const float*     x     = (const float*)d_in[0];
    const long long* edge  = (const long long*)d_in[1];   // [2, E] int64
    const long long* batch = (const long long*)d_in[2];   // [N]    int64
    const float*     W1    = (const float*)d_in[3];
    const float*     b1    = (const float*)d_in[4];
    const float*     W2    = (const float*)d_in[5];
    const float*     b2    = (const float*)d_in[6];
    const float*     Wlin  = (const float*)d_in[7];
    const float*     blin  = (const float*)d_in[8];
    float*           out   = (float*)d_out;

    const int N = in_sizes[0] / 64;      // 100000
    const int E = in_sizes[1] / 2;       // 1200000
    const long long EF = (long long)E * 64;

    // Workspace layout (floats)
    float* ws     = (float*)d_ws;
    float* dinv   = ws;                              // [N]  (deg -> dinv)
    float* invdeg = ws + (size_t)N;                  // [N]
    float* bufA   = ws + 2 * (size_t)N;              // [N*64]
    float* bufB   = bufA + (size_t)N * 64;           // [N*64]
    float* pooled = bufB + (size_t)N * 64;           // [512*64]

    const long long* src = edge;
    const long long* dst = edge + E;

    hipMemsetAsync(dinv,   0, (size_t)N * sizeof(float), stream);
    hipMemsetAsync(pooled, 0, (size_t)NUM_GRAPHS * 64 * sizeof(float), stream);

    deg_count_kernel<<<(E + 255) / 256, 256, 0, stream>>>(dst, dinv, E);
    deg_finalize_kernel<<<(N + 255) / 256, 256, 0, stream>>>(dinv, invdeg, N);

    const unsigned gScatter = (unsigned)((EF + 255) / 256);
    const int      nvec     = N * 16;   // float4 items

    // ---- Layer 1:  h1raw = x @ W1 ; B1 = scatter + selfloop -------------
    gemm64_wmma_kernel<false><<<N / 16, 128, 0, stream>>>(x, W1, nullptr, bufA);
    selfloop_init_kernel<<<(nvec + 255) / 256, 256, 0, stream>>>(bufA, invdeg, bufB, nvec);
    scatter_kernel<<<gScatter, 256, 0, stream>>>(bufA, src, dst, dinv, bufB, EF);

    // ---- Layer 2: A2 = relu(B1 + b1) @ W2  (epilogue fused into A read) -
    gemm64_wmma_kernel<true><<<N / 16, 128, 0, stream>>>(bufB, W2, b1, bufA);
    selfloop_init_kernel<<<(nvec + 255) / 256, 256, 0, stream>>>(bufA, invdeg, bufB, nvec);
    scatter_kernel<<<gScatter, 256, 0, stream>>>(bufA, src, dst, dinv, bufB, EF);

    // ---- Pool (fused bias+ReLU) + head ----------------------------------
    pool_kernel<<<(N * 64 + 255) / 256, 256, 0, stream>>>(bufB, b2, batch, pooled, N * 64);
    final_linear_kernel<<<(NUM_GRAPHS * 2 + 255) / 256, 256, 0, stream>>>(
        pooled, Wlin, blin, out, NUM_GRAPHS * 2);
}